// Warp_20993800143227
// MI455X (gfx1250) — compile-verified
//
#include <hip/hip_runtime.h>
#include <stdint.h>

#define K_PTS 50
#define NV    50000
#define B_BATCH 64
#define PPT   4          // points per thread
#define BLOCK 256

// Convert a generic pointer to a __shared__ object into a raw LDS byte offset
// (AS(3) pointers are 32-bit LDS byte offsets on AMDGPU).
__device__ __forceinline__ unsigned lds_offset_of(const void* p) {
  return (unsigned)(unsigned long long)
         (__attribute__((address_space(3))) const char*)p;
}

__global__ __launch_bounds__(BLOCK) void Warp_20993800143227_kernel(
    const float* __restrict__ vertices,      // [N,3]
    const float* __restrict__ warp_params,   // [B,K]
    const float* __restrict__ keypoints,     // [K,3]
    const float* __restrict__ warp_vectors,  // [K,3]
    float* __restrict__ out)                 // [B,N,3]
{
  // Per k (32B stride): kpx kpy kpz wvx | wvy wvz p pad
  __shared__ __align__(16) float smem[K_PTS * 8];

  const int tid = threadIdx.x;
  const int b   = blockIdx.y;

  // ---- Stage constants into LDS with CDNA5 async global->LDS loads ----
  for (int i = tid; i < 350; i += BLOCK) {
    unsigned ldsOff;
    const float* gsrc;
    if (i < 150) {                     // keypoints
      int k = i / 3, c = i - 3 * k;
      ldsOff = (unsigned)(k * 32 + c * 4);
      gsrc = keypoints + i;
    } else if (i < 300) {              // warp_vectors
      int j = i - 150;
      int k = j / 3, c = j - 3 * k;
      ldsOff = (unsigned)(k * 32 + 12 + c * 4);
      gsrc = warp_vectors + j;
    } else {                           // this block's warp_params row
      int k = i - 300;
      ldsOff = (unsigned)(k * 32 + 24);
      gsrc = warp_params + b * K_PTS + k;
    }
    unsigned dst = lds_offset_of(smem) + ldsOff;
    unsigned long long ga = (unsigned long long)(uintptr_t)gsrc;
    asm volatile("global_load_async_to_lds_b32 %0, %1, off"
                 :: "v"(dst), "v"(ga)
                 : "memory");
  }
  asm volatile("s_wait_asynccnt 0x0" ::: "memory");
  __syncthreads();

  const int j = blockIdx.x * BLOCK + tid;   // point-quad index
  if (j < NV / PPT) {
    // 12 consecutive floats = 3 coalesced float4 loads (48B aligned)
    const float4* vin = (const float4*)vertices;
    float4 f0 = vin[j * 3 + 0];
    float4 f1 = vin[j * 3 + 1];
    float4 f2 = vin[j * 3 + 2];
    float vx[PPT] = {f0.x, f0.w, f1.z, f2.y};
    float vy[PPT] = {f0.y, f1.x, f1.w, f2.z};
    float vz[PPT] = {f0.z, f1.y, f2.x, f2.w};

    const float C = -2.16404256133345f;     // -1.5 * log2(e)

    for (int k = 0; k < K_PTS; ++k) {
      // two uniform ds_load_b128 per step
      float4 c0 = *(const float4*)&smem[k * 8];
      float4 c1 = *(const float4*)&smem[k * 8 + 4];
      const float kx = c0.x, ky = c0.y, kz = c0.z;
      const float wx = c0.w, wy = c1.x, wz = c1.y;
      const float p  = c1.z;
#pragma unroll
      for (int i = 0; i < PPT; ++i) {
        float dx = vx[i] - kx;
        float dy = vy[i] - ky;
        float dz = vz[i] - kz;
        float d2 = fmaf(dz, dz, fmaf(dy, dy, dx * dx));
        float s  = p * __builtin_amdgcn_exp2f(d2 * C);   // exp(-1.5*d2)
        vx[i] = fmaf(s, wx, vx[i]);
        vy[i] = fmaf(s, wy, vy[i]);
        vz[i] = fmaf(s, wz, vz[i]);
      }
    }

    float4 o0 = {vx[0], vy[0], vz[0], vx[1]};
    float4 o1 = {vy[1], vz[1], vx[2], vy[2]};
    float4 o2 = {vz[2], vx[3], vy[3], vz[3]};
    float4* op = (float4*)(out + (size_t)b * (NV * 3));
    op[j * 3 + 0] = o0;
    op[j * 3 + 1] = o1;
    op[j * 3 + 2] = o2;
  }
}

extern "C" void kernel_launch(void* const* d_in, const int* in_sizes, int n_in,
                              void* d_out, int out_size, void* d_ws, size_t ws_size,
                              hipStream_t stream) {
  const float* vertices     = (const float*)d_in[0];  // 150000
  const float* warp_params  = (const float*)d_in[1];  // 64*50
  const float* keypoints    = (const float*)d_in[2];  // 150
  const float* warp_vectors = (const float*)d_in[3];  // 150
  float* out = (float*)d_out;                         // 64*50000*3

  dim3 grid((NV / PPT + BLOCK - 1) / BLOCK, B_BATCH); // (49, 64)
  Warp_20993800143227_kernel<<<grid, dim3(BLOCK), 0, stream>>>(
      vertices, warp_params, keypoints, warp_vectors, out);
}